// Attention_1812476199083
// MI455X (gfx1250) — compile-verified
//
#include <hip/hip_runtime.h>
#include <math.h>

typedef float v2f __attribute__((ext_vector_type(2)));
typedef float v8f __attribute__((ext_vector_type(8)));

#define BATCH 4
#define CDIM  64
#define HH    256
#define WWID  256
#define WP    129            // rfft width bins
#define NPOS  (HH*WP)        // 33024
#define NHEAD 8
#define CPH   32             // channels per head after FACTOR split
#define COLS  (NPOS/2)       // 16512
#define PIX   (HH*WWID)      // 65536
#define KSPLIT 32
#define KCHUNK (COLS/KSPLIT) // 516 (multiple of 4)

// ---------------- WMMA helper: D = A(16x4,f32) x B(4x16,f32) + C ----------------
static __device__ __forceinline__ v8f wmma4(v2f a, v2f b, v8f c) {
  return __builtin_amdgcn_wmma_f32_16x16x4_f32(false, a, false, b, (short)0, c,
                                               false, false);
}

// ---------------- permutation (stable argsort of mult(row)*col) -----------------
__global__ void k_rank(int* __restrict__ idx, int* __restrict__ inv) {
  int i = blockIdx.x * blockDim.x + threadIdx.x;
  if (i >= NPOS) return;
  int r = i / WP, ccol = i % WP;
  int mult = r < 128 ? r : 255 - r;
  int v = mult * ccol;
  int rank = 0;
  for (int rr = 0; rr < 256; ++rr) {
    int mr = rr < 128 ? rr : 255 - rr;
    if (v > 0) {
      if (mr == 0) rank += WP;                          // all 129 cols are 0 < v
      else { int q = (v - 1) / mr + 1; rank += q > WP ? WP : q; }
      if (mr > 0 && v % mr == 0) {                      // equal values, earlier idx
        int cq = v / mr;
        if (cq < WP && rr * WP + cq < i) rank += 1;
      }
    } else {                                            // v == 0
      if (mr == 0) {
        int base = rr * WP;
        if (base + WP <= i) rank += WP;
        else if (base < i)  rank += i - base;
      } else if (rr * WP < i) rank += 1;                // col 0 of row rr
    }
  }
  inv[i] = rank;
  idx[rank] = i;
}

__global__ void k_zero(float* __restrict__ p, int n) {
  int i = blockIdx.x * blockDim.x + threadIdx.x;
  if (i < n) p[i] = 0.f;
}

// ---------------- qkv 1x1 conv: (192x64) x (64 x 65536) per batch ---------------
__global__ void k_qkv(const float* __restrict__ x, const float* __restrict__ wqkv,
                      float* __restrict__ out) {
  int lane = threadIdx.x & 31;
  int wid  = (blockIdx.x * blockDim.x + threadIdx.x) >> 5;
  int lo = lane & 15, half = lane >> 4;
  int ntile = wid & 4095;            // 4096 pixel tiles
  int t  = wid >> 12;                // 0..47
  int m0 = (t % 12) << 4;
  int b  = t / 12;
  int n0 = ntile << 4;
  const float* xb = x + (size_t)b * CDIM * PIX;
  v8f acc = {0.f,0.f,0.f,0.f,0.f,0.f,0.f,0.f};
  for (int k = 0; k < CDIM; k += 4) {
    v2f a, bb;
    a.x  = wqkv[(m0 + lo) * CDIM + k + 2*half + 0];
    a.y  = wqkv[(m0 + lo) * CDIM + k + 2*half + 1];
    bb.x = xb[(size_t)(k + 2*half + 0) * PIX + n0 + lo];
    bb.y = xb[(size_t)(k + 2*half + 1) * PIX + n0 + lo];
    if (k + 4 < CDIM)
      __builtin_prefetch(&xb[(size_t)(k + 4 + 2*half) * PIX + n0 + lo], 0, 1);
    acc = wmma4(a, bb, acc);
  }
  float* ob = out + ((size_t)b * 192 + m0) * PIX + n0;
  for (int v = 0; v < 8; ++v)
    ob[(size_t)(8*half + v) * PIX + lo] = acc[v];
}

// ---------------- depthwise 3x3 SAME ----------------
__global__ void k_dwconv(const float* __restrict__ in, const float* __restrict__ wdw,
                         float* __restrict__ out) {
  size_t i = (size_t)blockIdx.x * blockDim.x + threadIdx.x; // 4*192*65536 total
  int wpos = (int)(i & 255);
  int hpos = (int)((i >> 8) & 255);
  int ch   = (int)((i >> 16) % 192);
  const float* w9  = wdw + ch * 9;
  const float* base = in + ((i >> 16) << 16);
  float s = 0.f;
  for (int dy = -1; dy <= 1; ++dy) {
    int hy = hpos + dy; if (hy < 0 || hy > 255) continue;
    for (int dx = -1; dx <= 1; ++dx) {
      int wx = wpos + dx; if (wx < 0 || wx > 255) continue;
      s += base[(hy << 8) + wx] * w9[(dy + 1) * 3 + dx + 1];
    }
  }
  out[i] = s;
}

// ---------------- radix-2 DIT 256-pt FFT in LDS (one block per transform) -------
static __device__ __forceinline__ void fft256(float* re, float* im, int t, float sgn) {
  const float TWO_PI = 6.283185307179586f;
  for (int s = 1; s <= 8; ++s) {
    int mlen = 1 << s;
    int hl = mlen >> 1;
    if (t < 128) {
      int grp = t >> (s - 1);
      int pos = t & (hl - 1);
      int i0 = (grp << s) + pos;
      int i1 = i0 + hl;
      float ang = sgn * TWO_PI * (float)pos / (float)mlen;
      float si, co;
      __sincosf(ang, &si, &co);
      float tr = co * re[i1] - si * im[i1];
      float ti = co * im[i1] + si * re[i1];
      re[i1] = re[i0] - tr; im[i1] = im[i0] - ti;
      re[i0] += tr;         im[i0] += ti;
    }
    __syncthreads();
  }
}

// rfft along w: real 256 -> 129 bins, scaled 1/16 (half of the 2D ortho norm)
__global__ void k_rowfft(const float* __restrict__ in, float* __restrict__ cre,
                         float* __restrict__ cim, int chOff) {
  __shared__ float re[256], im[256];
  int t = threadIdx.x;
  int row = blockIdx.x;            // (b*64+c)*256 + h
  int bc = row >> 8;
  int b = bc >> 6, c = bc & 63;
  const float* src = in + (((size_t)b * 192 + chOff + c) * 256 + (row & 255)) * 256;
  int rv = __brev((unsigned)t) >> 24;
  re[rv] = src[t]; im[rv] = 0.f;
  __syncthreads();
  fft256(re, im, t, -1.f);
  if (t < WP) {
    cre[(size_t)row * WP + t] = re[t] * 0.0625f;
    cim[(size_t)row * WP + t] = im[t] * 0.0625f;
  }
}

// fft along h (complex) + scatter directly into head-packed layout via inv[]
__global__ void k_colfft_pack(const float* __restrict__ cre, const float* __restrict__ cim,
                              const int* __restrict__ inv, float* __restrict__ packed) {
  __shared__ float re[256], im[256];
  int t = threadIdx.x;
  int col = blockIdx.x;            // (b*64+c)*129 + w
  int w = col % WP;
  int bc = col / WP;
  int b = bc >> 6, c = bc & 63;
  size_t base = (size_t)bc * 256 * WP + w;
  int rv = __brev((unsigned)t) >> 24;
  re[rv] = cre[base + (size_t)t * WP];
  im[rv] = cim[base + (size_t)t * WP];
  __syncthreads();
  fft256(re, im, t, -1.f);
  int j = t * WP + w;              // original flat position (h*129+w)
  int p = inv[j];                  // permuted position
  int f = p / COLS, ccol = p % COLS;
  float vr = re[t] * 0.0625f, vi = im[t] * 0.0625f;
  int hd = c >> 4, cc = c & 15;    // real channel rc=c
  packed[(((size_t)b * NHEAD + hd) * CPH + cc * 2 + f) * COLS + ccol] = vr;
  int c2 = 64 + c; int hd2 = c2 >> 4, cc2 = c2 & 15;   // imag channel rc=64+c
  packed[(((size_t)b * NHEAD + hd2) * CPH + cc2 * 2 + f) * COLS + ccol] = vi;
}

// ---------------- per-row L2 scale (1/max(||row||,1e-12)) ----------------
__global__ void k_norms(const float* __restrict__ pk, float* __restrict__ scale) {
  __shared__ float sm[256];
  int row = blockIdx.x;            // b*256 + (h*32+c')
  const float* src = pk + (size_t)row * COLS;
  float s = 0.f;
  for (int i = threadIdx.x; i < COLS; i += 256) { float v = src[i]; s += v * v; }
  sm[threadIdx.x] = s; __syncthreads();
  for (int st = 128; st > 0; st >>= 1) {
    if (threadIdx.x < st) sm[threadIdx.x] += sm[threadIdx.x + st];
    __syncthreads();
  }
  if (threadIdx.x == 0) {
    float n = sqrtf(sm[0]); n = n < 1e-12f ? 1e-12f : n;
    scale[row] = 1.0f / n;
  }
}

// ---------------- scores: 32x32 <- qh(32xCOLS) x kh^T, split-K + atomics --------
__global__ void k_attn(const float* __restrict__ qp, const float* __restrict__ kp,
                       float* __restrict__ attn) {
  int lane = threadIdx.x & 31;
  int wid  = (blockIdx.x * blockDim.x + threadIdx.x) >> 5; // 0..4095
  int lo = lane & 15, half = lane >> 4;
  int ks   = wid & (KSPLIT - 1);
  int tile = wid >> 5;             // 0..127
  int n0 = (tile & 1) << 4;
  int m0 = ((tile >> 1) & 1) << 4;
  int bh = tile >> 2;              // b*8+h
  const float* qb = qp + (size_t)bh * CPH * COLS;
  const float* kb = kp + (size_t)bh * CPH * COLS;
  int k0 = ks * KCHUNK;
  v8f acc = {0.f,0.f,0.f,0.f,0.f,0.f,0.f,0.f};
  for (int k = k0; k < k0 + KCHUNK; k += 4) {
    v2f a, bb;
    a.x  = qb[(size_t)(m0 + lo) * COLS + k + 2*half + 0];
    a.y  = qb[(size_t)(m0 + lo) * COLS + k + 2*half + 1];
    bb.x = kb[(size_t)(n0 + lo) * COLS + k + 2*half + 0];
    bb.y = kb[(size_t)(n0 + lo) * COLS + k + 2*half + 1];
    __builtin_prefetch(&qb[(size_t)(m0 + lo) * COLS + k + 16], 0, 1);
    acc = wmma4(a, bb, acc);
  }
  float* ab = attn + (size_t)bh * CPH * CPH;
  for (int v = 0; v < 8; ++v)
    atomicAdd(&ab[(m0 + 8*half + v) * CPH + n0 + lo], acc[v]);
}

// ---------------- fixup: attn = softmax1(raw * sq * sk * temp[h]) ---------------
__global__ void k_softmax(float* __restrict__ attn, const float* __restrict__ qs,
                          const float* __restrict__ ksc, const float* __restrict__ temp) {
  int bh = blockIdx.x;             // 0..31
  int h = bh & 7;
  int r = threadIdx.x;             // 0..31
  float* row = attn + ((size_t)bh * CPH + r) * CPH;
  float qfac = qs[bh * CPH + r] * temp[h];
  float sum = 0.f;
  for (int d = 0; d < CPH; ++d)
    sum += expf(row[d] * qfac * ksc[bh * CPH + d]);
  float invs = 1.0f / (sum + 1.0f);
  for (int d = 0; d < CPH; ++d)
    row[d] = expf(row[d] * qfac * ksc[bh * CPH + d]) * invs;
}

// ---------------- o = attn(32x32) x vh(32xCOLS), scatter into complex buf -------
__global__ void k_ogemm(const float* __restrict__ attn, const float* __restrict__ vp,
                        const int* __restrict__ idx, float* __restrict__ cre,
                        float* __restrict__ cim) {
  int lane = threadIdx.x & 31;
  int wid  = (blockIdx.x * blockDim.x + threadIdx.x) >> 5; // 0..66047
  int lo = lane & 15, half = lane >> 4;
  int ntile = wid % 1032;          // COLS/16
  int t = wid / 1032;              // 0..63
  int m0 = (t & 1) << 4;
  int bh = t >> 1;                 // b*8+h
  int b = bh >> 3, h = bh & 7;
  int n0 = ntile << 4;
  const float* ab = attn + (size_t)bh * CPH * CPH;
  const float* vb = vp + (size_t)bh * CPH * COLS;
  v8f acc = {0.f,0.f,0.f,0.f,0.f,0.f,0.f,0.f};
  for (int k = 0; k < CPH; k += 4) {
    v2f a, bb;
    a.x  = ab[(m0 + lo) * CPH + k + 2*half + 0];
    a.y  = ab[(m0 + lo) * CPH + k + 2*half + 1];
    bb.x = vb[(size_t)(k + 2*half + 0) * COLS + n0 + lo];
    bb.y = vb[(size_t)(k + 2*half + 1) * COLS + n0 + lo];
    acc = wmma4(a, bb, acc);
  }
  for (int v = 0; v < 8; ++v) {
    int cp = m0 + 8*half + v;      // c' in 0..31
    int f = cp & 1, cc = cp >> 1;
    int rc = h * 16 + cc;          // r-channel 0..127
    int col = n0 + lo;
    int j = idx[f * COLS + col];   // back to original (h*129+w) position
    int hrow = j / WP, wq = j % WP;
    size_t o = (((size_t)b * 64 + (rc & 63)) * 256 + hrow) * WP + wq;
    if (rc < 64) cre[o] = acc[v]; else cim[o] = acc[v];
  }
}

// ---------------- inverse fft along h (in place), scale 1/16 --------------------
__global__ void k_colifft(float* __restrict__ cre, float* __restrict__ cim) {
  __shared__ float re[256], im[256];
  int t = threadIdx.x;
  int col = blockIdx.x;            // (b*64+c)*129 + w
  int w = col % WP;
  int bc = col / WP;
  size_t base = (size_t)bc * 256 * WP + w;
  int rv = __brev((unsigned)t) >> 24;
  re[rv] = cre[base + (size_t)t * WP];
  im[rv] = cim[base + (size_t)t * WP];
  __syncthreads();
  fft256(re, im, t, +1.f);
  cre[base + (size_t)t * WP] = re[t] * 0.0625f;
  cim[base + (size_t)t * WP] = im[t] * 0.0625f;
}

// ---------------- irfft along w: 129 bins -> 256 real, scale 1/16 ---------------
__global__ void k_rowirfft(const float* __restrict__ cre, const float* __restrict__ cim,
                           float* __restrict__ out) {
  __shared__ float re[256], im[256];
  int t = threadIdx.x;
  int row = blockIdx.x;            // (b*64+c)*256 + h
  size_t base = (size_t)row * WP;
  int rv = __brev((unsigned)t) >> 24;
  float xr, xi;
  if (t <= 128) { xr = cre[base + t];        xi =  cim[base + t]; }
  else          { xr = cre[base + 256 - t];  xi = -cim[base + 256 - t]; }
  re[rv] = xr; im[rv] = xi;
  __syncthreads();
  fft256(re, im, t, +1.f);
  out[(size_t)row * 256 + t] = re[t] * 0.0625f;
}

// ---------------- proj 1x1 conv: (64x64) x (64 x 65536) per batch ---------------
__global__ void k_proj(const float* __restrict__ in, const float* __restrict__ wp,
                       float* __restrict__ out) {
  int lane = threadIdx.x & 31;
  int wid  = (blockIdx.x * blockDim.x + threadIdx.x) >> 5;
  int lo = lane & 15, half = lane >> 4;
  int ntile = wid & 4095;
  int t = wid >> 12;               // 0..15
  int m0 = (t & 3) << 4;
  int b  = t >> 2;
  int n0 = ntile << 4;
  const float* ib = in + (size_t)b * CDIM * PIX;
  v8f acc = {0.f,0.f,0.f,0.f,0.f,0.f,0.f,0.f};
  for (int k = 0; k < CDIM; k += 4) {
    v2f a, bb;
    a.x  = wp[(m0 + lo) * CDIM + k + 2*half + 0];
    a.y  = wp[(m0 + lo) * CDIM + k + 2*half + 1];
    bb.x = ib[(size_t)(k + 2*half + 0) * PIX + n0 + lo];
    bb.y = ib[(size_t)(k + 2*half + 1) * PIX + n0 + lo];
    if (k + 4 < CDIM)
      __builtin_prefetch(&ib[(size_t)(k + 4 + 2*half) * PIX + n0 + lo], 0, 1);
    acc = wmma4(a, bb, acc);
  }
  float* ob = out + ((size_t)b * CDIM + m0) * PIX + n0;
  for (int v = 0; v < 8; ++v)
    ob[(size_t)(8*half + v) * PIX + lo] = acc[v];
}

extern "C" void kernel_launch(void* const* d_in, const int* in_sizes, int n_in,
                              void* d_out, int out_size, void* d_ws, size_t ws_size,
                              hipStream_t stream) {
  const float* x     = (const float*)d_in[0];
  const float* wqkv  = (const float*)d_in[1];
  const float* wdw   = (const float*)d_in[2];
  const float* wproj = (const float*)d_in[3];
  const float* temp  = (const float*)d_in[4];

  char* ws = (char*)d_ws;
  size_t off = 0;
  int* idx = (int*)(ws + off); off += (size_t)NPOS * sizeof(int);
  int* inv = (int*)(ws + off); off += (size_t)NPOS * sizeof(int);
  off = (off + 255) & ~(size_t)255;
  float* qkv = (float*)(ws + off); off += (size_t)BATCH * 192 * PIX * 4;
  float* dwb = (float*)(ws + off); off += (size_t)BATCH * 192 * PIX * 4;
  float* cre = (float*)(ws + off); off += (size_t)BATCH * 64 * 256 * WP * 4;
  float* cim = (float*)(ws + off); off += (size_t)BATCH * 64 * 256 * WP * 4;
  float* qp  = (float*)(ws + off); off += (size_t)BATCH * 256 * COLS * 4;
  float* kp  = (float*)(ws + off); off += (size_t)BATCH * 256 * COLS * 4;
  float* vp  = (float*)(ws + off); off += (size_t)BATCH * 256 * COLS * 4;
  float* att = (float*)(ws + off); off += (size_t)BATCH * NHEAD * CPH * CPH * 4;
  float* qs  = (float*)(ws + off); off += (size_t)BATCH * 256 * 4;
  float* ksn = (float*)(ws + off); off += (size_t)BATCH * 256 * 4;
  float* realbuf = qp;  // qp free after the score GEMM; reuse for irfft output

  k_rank<<<NPOS / 256, 256, 0, stream>>>(idx, inv);
  k_zero<<<(BATCH * NHEAD * CPH * CPH + 255) / 256, 256, 0, stream>>>(
      att, BATCH * NHEAD * CPH * CPH);
  k_qkv<<<24576, 256, 0, stream>>>(x, wqkv, qkv);
  k_dwconv<<<196608, 256, 0, stream>>>(qkv, wdw, dwb);
  float* packs[3] = {qp, kp, vp};
  for (int tq = 0; tq < 3; ++tq) {
    k_rowfft<<<BATCH * 64 * 256, 256, 0, stream>>>(dwb, cre, cim, tq * 64);
    k_colfft_pack<<<BATCH * 64 * WP, 256, 0, stream>>>(cre, cim, inv, packs[tq]);
  }
  k_norms<<<BATCH * 256, 256, 0, stream>>>(qp, qs);
  k_norms<<<BATCH * 256, 256, 0, stream>>>(kp, ksn);
  k_attn<<<512, 256, 0, stream>>>(qp, kp, att);
  k_softmax<<<BATCH * NHEAD, CPH, 0, stream>>>(att, qs, ksn, temp);
  k_ogemm<<<8256, 256, 0, stream>>>(att, vp, idx, cre, cim);
  k_colifft<<<BATCH * 64 * WP, 256, 0, stream>>>(cre, cim);
  k_rowirfft<<<BATCH * 64 * 256, 256, 0, stream>>>(cre, cim, realbuf);
  k_proj<<<8192, 256, 0, stream>>>(realbuf, wproj, (float*)d_out);
}